// ChamferLoss_13606456393966
// MI455X (gfx1250) — compile-verified
//
#include <hip/hip_runtime.h>

// CDNA5 / gfx1250 Chamfer distance.
// dist(m,n) = ||p||^2 + ||q||^2 - 2 p.q computed entirely inside
// V_WMMA_F32_16X16X4_F32 with A=(px,py,pz,1), B=(-2qx,-2qy,-2qz,||q||^2), C=||p||^2.
// The staged LDS layout doubles as the exact per-lane-half B operand:
//   float4 (-2x,-2y,-2z,qsq) == { (K0,K1) for lanes 0-15, (K2,K3) for lanes 16-31 }
// so the hot loop is just: ds_load_b64 -> v_wmma -> 8x v_min per 256 pairs.

typedef __attribute__((ext_vector_type(2))) float v2f;
typedef __attribute__((ext_vector_type(8))) float v8f;

#define BLOCK          256
#define WAVES_PER_BLK  8           // 256 / wave32
#define STRIP          16          // pred rows handled per wave (one WMMA M-tile)
#define PBLOCK         (WAVES_PER_BLK * STRIP)   // 128 pred points per block
#define QCHUNK         512         // gt points staged in LDS per iteration

__global__ __launch_bounds__(BLOCK)
void chamfer_min_pass(const float* __restrict__ P, const float* __restrict__ Q,
                      int NP, int NQ, float scale, float* __restrict__ partial)
{
    __shared__ float4 qs[QCHUNK];        // staged gt: (-2x, -2y, -2z, ||q||^2)
    __shared__ float  psq[PBLOCK];       // ||p||^2 for this block's points
    __shared__ float  wsum[WAVES_PER_BLK];

    const int blocksPerBatch = NP / PBLOCK;
    const int b     = blockIdx.x / blocksPerBatch;
    const int pb    = blockIdx.x % blocksPerBatch;
    const float* Pb = P + (size_t)b * NP * 3;
    const float* Qb = Q + (size_t)b * NQ * 3;
    const int pbase = pb * PBLOCK;

    const int tid  = threadIdx.x;
    const int wave = tid >> 5;
    const int lane = tid & 31;
    const int half = lane >> 4;          // 0: lanes 0-15, 1: lanes 16-31
    const int l16  = lane & 15;

    // Stage ||p||^2 for the block's 128 pred points.
    if (tid < PBLOCK) {
        int m   = pbase + tid;
        float x = Pb[m * 3 + 0], y = Pb[m * 3 + 1], z = Pb[m * 3 + 2];
        psq[tid] = __builtin_fmaf(x, x, __builtin_fmaf(y, y, z * z));
    }
    __syncthreads();

    // A matrix (16x4 f32): v0 holds K0 (lanes 0-15) / K2 (lanes 16-31),
    //                      v1 holds K1 / K3. Row m = lane%16.
    const int m = pbase + wave * STRIP + l16;
    const float px = Pb[m * 3 + 0], py = Pb[m * 3 + 1], pz = Pb[m * 3 + 2];
    v2f A;
    A.x = half ? pz   : px;     // K2 : K0
    A.y = half ? 1.0f : py;     // K3 : K1

    // C matrix: C[i] = ||p||^2 of row (i + 8*half) of this wave's strip.
    v8f C;
#pragma unroll
    for (int i = 0; i < 8; ++i)
        C[i] = psq[wave * STRIP + i + 8 * half];

    float rmin[8];
#pragma unroll
    for (int i = 0; i < 8; ++i) rmin[i] = 3.402823466e+38f;

    // Per-lane B-operand view of the staged chunk: element (2*n + half) is
    // exactly this lane-half's (K pair) for gt point n. ds_load_b64, no selects.
    const float2* qs2 = reinterpret_cast<const float2*>(qs);

    for (int qc = 0; qc < NQ; qc += QCHUNK) {
        // Cooperative stage of QCHUNK gt points, pre-transformed.
#pragma unroll
        for (int r = 0; r < QCHUNK / BLOCK; ++r) {
            int j = r * BLOCK + tid;
            int n = qc + j;
            float x = Qb[n * 3 + 0], y = Qb[n * 3 + 1], z = Qb[n * 3 + 2];
            float4 v;
            v.x = -2.0f * x;
            v.y = -2.0f * y;
            v.z = -2.0f * z;
            v.w = __builtin_fmaf(x, x, __builtin_fmaf(y, y, z * z));
            qs[j] = v;
        }
        __syncthreads();

        // 32 WMMA tiles per chunk. Per tile: 1 ds_load_b64,
        // 1 v_wmma_f32_16x16x4_f32, 8 v_min (fused to min3) — 256 pairs.
#pragma unroll 8
        for (int t = 0; t < QCHUNK / 16; ++t) {
            float2 bq = qs2[((t * 16 + l16) << 1) + half];
            v2f Bm;
            Bm.x = bq.x;
            Bm.y = bq.y;
            v8f D = __builtin_amdgcn_wmma_f32_16x16x4_f32(
                false, A, false, Bm, (short)0, C, false, false);
#pragma unroll
            for (int i = 0; i < 8; ++i)
                rmin[i] = fminf(rmin[i], D[i]);
        }
        __syncthreads();
    }

    // Min across the 16 lanes of each half (D layout: n = lane%16, so xor
    // masks 1,2,4,8 stay inside a half). After this every lane of a half
    // holds the full min for all 8 of its rows.
#pragma unroll
    for (int s = 1; s < 16; s <<= 1) {
#pragma unroll
        for (int i = 0; i < 8; ++i)
            rmin[i] = fminf(rmin[i], __shfl_xor(rmin[i], s, 32));
    }

    // Clamp (commutes with min) and sum the 8 rows of this half; lanes within
    // a half are identical, shfl_xor(16) adds the other half exactly once.
    float ssum = 0.0f;
#pragma unroll
    for (int i = 0; i < 8; ++i) ssum += fmaxf(rmin[i], 0.0f);
    ssum += __shfl_xor(ssum, 16, 32);
    if (lane == 0) wsum[wave] = ssum * scale;
    __syncthreads();

    if (tid == 0) {
        float s = 0.0f;
#pragma unroll
        for (int w = 0; w < WAVES_PER_BLK; ++w) s += wsum[w];
        partial[blockIdx.x] = s;
    }
}

__global__ __launch_bounds__(256)
void reduce_partials(const float* __restrict__ partial, int count,
                     float* __restrict__ out)
{
    __shared__ float sh[256];
    float s = 0.0f;
    for (int i = threadIdx.x; i < count; i += 256) s += partial[i];
    sh[threadIdx.x] = s;
    __syncthreads();
    for (int off = 128; off > 0; off >>= 1) {
        if (threadIdx.x < off) sh[threadIdx.x] += sh[threadIdx.x + off];
        __syncthreads();
    }
    if (threadIdx.x == 0) out[0] = sh[0];
}

extern "C" void kernel_launch(void* const* d_in, const int* in_sizes, int n_in,
                              void* d_out, int out_size, void* d_ws, size_t ws_size,
                              hipStream_t stream) {
    const float* pred = (const float*)d_in[0];   // [B, N, 3] f32
    const float* gt   = (const float*)d_in[1];   // [B, M, 3] f32
    const int B = 8;                             // per reference setup
    const int N = in_sizes[0] / (B * 3);
    const int M = in_sizes[1] / (B * 3);

    float* ws = (float*)d_ws;
    const int blocks1 = B * (N / PBLOCK);
    const int blocks2 = B * (M / PBLOCK);
    const float s1 = 1.0f / ((float)B * (float)N);
    const float s2 = 1.0f / ((float)B * (float)M);

    // pred -> gt mins
    chamfer_min_pass<<<blocks1, BLOCK, 0, stream>>>(pred, gt, N, M, s1, ws);
    // gt -> pred mins
    chamfer_min_pass<<<blocks2, BLOCK, 0, stream>>>(gt, pred, M, N, s2, ws + blocks1);
    // final scalar
    reduce_partials<<<1, 256, 0, stream>>>(ws, blocks1 + blocks2, (float*)d_out);
}